// MGCBR_6502580486179
// MI455X (gfx1250) — compile-verified
//
#include <hip/hip_runtime.h>

#define N_NODES 65536
#define D_FEAT  64
#define DEG     16
#define ALPHA   0.2f
#define GAT_EPS 1e-12f

typedef float v2f __attribute__((ext_vector_type(2)));
typedef float v8f __attribute__((ext_vector_type(8)));

// ---------------------------------------------------------------------------
// Kernel 0 (tiny): u1 = W @ a[:64], u2 = W @ a[64:], c1 = b.a1, c2 = b.a2
// ---------------------------------------------------------------------------
__global__ __launch_bounds__(64) void gat_prep(const float* __restrict__ W,
                                               const float* __restrict__ a,
                                               const float* __restrict__ bias,
                                               float* __restrict__ u) {
    int k = threadIdx.x;                       // 64 threads
    float acc1 = 0.f, acc2 = 0.f;
    #pragma unroll 8
    for (int n = 0; n < 64; ++n) {
        float w = W[k * 64 + n];
        acc1 += w * a[n];
        acc2 += w * a[64 + n];
    }
    u[k]      = acc1;
    u[64 + k] = acc2;
    if (k == 0) {
        float c1 = 0.f, c2 = 0.f;
        for (int n = 0; n < 64; ++n) { c1 += bias[n] * a[n]; c2 += bias[n] * a[64 + n]; }
        u[128] = c1;
        u[129] = c2;
    }
}

// ---------------------------------------------------------------------------
// Kernel 1: h = X @ W + bias  via V_WMMA_F32_16X16X4_F32, fused with
//           s1 = X.u1 + c1, s2 = X.u2 + c2 (per-node attention scores).
// 256 threads = 8 waves, each wave owns a 16-row x 64-col tile of h.
// Block covers 128 rows -> grid 512.
// ---------------------------------------------------------------------------
#define WT_STRIDE 66   // padded LDS stride (floats), keeps 8B alignment, skews banks

__global__ __launch_bounds__(256) void gat_project(
    const float* __restrict__ X, const float* __restrict__ W,
    const float* __restrict__ bias, const float* __restrict__ u,
    float* __restrict__ h, float* __restrict__ s1, float* __restrict__ s2) {

    __shared__ float Wt[64 * WT_STRIDE];      // W transposed: Wt[n][k]

    int t = threadIdx.x;
    // cooperative transposed load of W (coalesced on the global read)
    #pragma unroll
    for (int i = 0; i < 16; ++i) {
        int e = t + 256 * i;
        int k = e >> 6;
        int n = e & 63;
        Wt[n * WT_STRIDE + k] = W[e];
    }
    __syncthreads();

    const int wave = t >> 5;
    const int lane = t & 31;
    const int row_base = blockIdx.x * 128 + wave * 16;
    const int m  = lane & 15;                 // M index (A) == N index (B)
    const int kp = (lane >> 4) * 2;           // K sub-pair selected by lane half

    v8f acc[4] = {};                          // 4 x (16x16 f32 accumulator)

    const float* xrow = X + (size_t)(row_base + m) * D_FEAT;

    #pragma unroll
    for (int kk = 0; kk < 16; ++kk) {         // K = 64 in steps of 4
        v2f afr = *(const v2f*)(xrow + 4 * kk + kp);
        #pragma unroll
        for (int nt = 0; nt < 4; ++nt) {
            v2f bfr = *(const v2f*)(&Wt[(16 * nt + m) * WT_STRIDE + 4 * kk + kp]);
            acc[nt] = __builtin_amdgcn_wmma_f32_16x16x4_f32(
                false, afr, false, bfr, (short)0, acc[nt], false, false);
        }
    }

    // bias add + store h tile.  C/D layout: VGPR r -> M = r + 8*(lane>=16),
    // N = (lane&15) + 16*nt
    const int mbase = 8 * (lane >> 4);
    #pragma unroll
    for (int nt = 0; nt < 4; ++nt) {
        int ncol = 16 * nt + (lane & 15);
        float bn = bias[ncol];
        #pragma unroll
        for (int r = 0; r < 8; ++r) {
            h[(size_t)(row_base + mbase + r) * D_FEAT + ncol] = acc[nt][r] + bn;
        }
    }

    // fused attention scores: lane L handles row (L&15), half (L>>4) of columns
    {
        int row  = row_base + (lane & 15);
        int half = lane >> 4;
        const float* xr  = X + (size_t)row * D_FEAT + 32 * half;
        const float* u1p = u + 32 * half;
        const float* u2p = u + 64 + 32 * half;
        float p1 = 0.f, p2 = 0.f;
        #pragma unroll 8
        for (int c = 0; c < 32; ++c) {
            float xv = xr[c];
            p1 += xv * u1p[c];
            p2 += xv * u2p[c];
        }
        p1 += __shfl_xor(p1, 16, 32);
        p2 += __shfl_xor(p2, 16, 32);
        if (half == 0) {
            s1[row] = p1 + u[128];
            s2[row] = p2 + u[129];
        }
    }
}

// ---------------------------------------------------------------------------
// Kernel 2: per-node segment softmax (DEG=16, contiguous) + weighted gather.
// One wave32 per node; lanes mirror the 16 edges in both halves, each lane
// accumulates 2 output features (float2 gathers -> 256B coalesced per row,
// served from L2 since h is 16MB << 192MB).
// ---------------------------------------------------------------------------
__global__ __launch_bounds__(256) void gat_aggregate(
    const int* __restrict__ indices, const float* __restrict__ h,
    const float* __restrict__ s1, const float* __restrict__ s2,
    float* __restrict__ out) {

    int t    = threadIdx.x;
    int wave = t >> 5;
    int lane = t & 31;
    int node = blockIdx.x * 8 + wave;

    int j   = lane & 15;
    int col = indices[node * DEG + j];

    float e = s1[node] + s2[col];
    e = (e > 0.f) ? e : ALPHA * e;            // leaky_relu

    // max over the 16-edge group (duplicated across halves)
    float mx = e;
    #pragma unroll
    for (int off = 1; off < 16; off <<= 1)
        mx = fmaxf(mx, __shfl_xor(mx, off, 32));

    float ex  = __expf(e - mx);
    float den = ex;
    #pragma unroll
    for (int off = 1; off < 16; off <<= 1)
        den += __shfl_xor(den, off, 32);

    float attn = ex / (den + GAT_EPS);

    // out[node][2*lane .. 2*lane+1] = sum_j attn_j * h[col_j][...]
    float accx = 0.f, accy = 0.f;
    #pragma unroll
    for (int jj = 0; jj < 16; ++jj) {
        float aj = __shfl(attn, jj, 32);
        int   cj = __shfl(col,  jj, 32);
        v2f hv = *(const v2f*)(h + (size_t)cj * D_FEAT + 2 * lane);
        accx += aj * hv.x;
        accy += aj * hv.y;
    }
    v2f res;
    res.x = accx;
    res.y = accy;
    *(v2f*)(out + (size_t)node * D_FEAT + 2 * lane) = res;
}

// ---------------------------------------------------------------------------
// launch
// ---------------------------------------------------------------------------
extern "C" void kernel_launch(void* const* d_in, const int* in_sizes, int n_in,
                              void* d_out, int out_size, void* d_ws, size_t ws_size,
                              hipStream_t stream) {
    const float* X       = (const float*)d_in[0];   // [N, 64]
    const float* W       = (const float*)d_in[1];   // [64, 64]
    const float* a       = (const float*)d_in[2];   // [128, 1]
    const float* bias    = (const float*)d_in[3];   // [64]
    // d_in[4] = indptr (uniform degree, unused)
    const int*   indices = (const int*)d_in[5];     // [N*16]
    float* out = (float*)d_out;

    // workspace layout (floats): h [N*64] | s1 [N] | s2 [N] | u1[64] u2[64] c1 c2
    float* ws = (float*)d_ws;
    float* h  = ws;
    float* s1 = ws + (size_t)N_NODES * D_FEAT;
    float* s2 = s1 + N_NODES;
    float* u  = s2 + N_NODES;

    gat_prep<<<1, 64, 0, stream>>>(W, a, bias, u);
    gat_project<<<N_NODES / 128, 256, 0, stream>>>(X, W, bias, u, h, s1, s2);
    gat_aggregate<<<N_NODES / 8, 256, 0, stream>>>(indices, h, s1, s2, out);
}